// ProgressiveSparseLocalAttention_34857954575058
// MI455X (gfx1250) — compile-verified
//
#include <hip/hip_runtime.h>
#include <hip/hip_bf16.h>

// ---------------------------------------------------------------------------
// ProgressiveSparseLocalAttention for MI455X (gfx1250, wave32)
//
// Stage 1: Ef = Wf @ Ft + bf, Eg = Wg @ Ft_eps + bg  via V_WMMA_F32_16X16X4_F32.
//          512-thread blocks (16 waves = 16 M-tiles) share one B panel staged
//          in LDS (K-transposed, padded) -> 16x less global B traffic, B fetch
//          is one aligned ds_load_b64 per WMMA operand.
// Stage 2: 33-offset local attention with LDS halo tiling (VALU FMAs; the
//          contraction is a diagonal band, so WMMA would waste 15/16 of work).
// LDS staging uses GLOBAL_LOAD_ASYNC_TO_LDS_B32 (ASYNCcnt) when available.
// ---------------------------------------------------------------------------

typedef __attribute__((ext_vector_type(2))) float v2f;
typedef __attribute__((ext_vector_type(8))) float v8f;

#define BATCH 2
#define CIN   256
#define HH    64
#define WW    64
#define HW    (HH * WW)          // 4096
#define NOFF  33

// ---- async global->LDS copy helpers (gfx1250 path with portable fallback) --
#if defined(__HIP_DEVICE_COMPILE__) &&                                         \
    __has_builtin(__builtin_amdgcn_global_load_async_to_lds_b32) &&            \
    __has_builtin(__builtin_amdgcn_s_wait_asynccnt)
#define PSLA_ASYNC_LDS 1
typedef __attribute__((address_space(1))) int g_i32;
typedef __attribute__((address_space(3))) int l_i32;
#endif

__device__ __forceinline__ void lds_copy_b32(const float* g, float* l) {
#ifdef PSLA_ASYNC_LDS
  __builtin_amdgcn_global_load_async_to_lds_b32(
      (g_i32*)(unsigned long long)(uintptr_t)g,
      (l_i32*)(unsigned int)(uintptr_t)l,
      0, 0);
#else
  *l = *g;
#endif
}

__device__ __forceinline__ void lds_copy_fence() {
#ifdef PSLA_ASYNC_LDS
  __builtin_amdgcn_s_wait_asynccnt(0);
#endif
}

// ---------------------------------------------------------------------------
// Kernel 1: dual 1x1-conv GEMM with f32 WMMA.
// Block: 512 threads = 16 waves; wave w owns M-tile [16w, 16w+16).
// All waves share a 64-pixel N-group; B panels (K-chunk 32) staged in LDS.
// A (16x4, M x K): lane L (m = L%16) holds a[j] = W[m][2*(L/16)+j]
// B (4x16, K x N): lane L (n = L%16) holds b[j] = X[2*(L/16)+j][n]
// D (16x16):       lane L, reg r     holds D[r + 8*(L/16)][L%16]
// ---------------------------------------------------------------------------
#define KC    32                 // K-chunk staged per barrier
#define KPAD  34                 // padded k-stride (keeps b64 8B-aligned,
                                 // spreads banks: gcd(34,64)=2)

__global__ __launch_bounds__(512) void psla_conv1x1_wmma(
    const float* __restrict__ Ft, const float* __restrict__ Fte,
    const float* __restrict__ Wf, const float* __restrict__ bf,
    const float* __restrict__ Wg, const float* __restrict__ bg,
    float* __restrict__ Ef, float* __restrict__ Eg) {
  __shared__ float bshare[2 * 64 * KPAD];          // 17408 B

  const int lane = threadIdx.x & 31;
  const int wid  = threadIdx.x >> 5;               // 16 waves
  const int b    = blockIdx.x >> 6;                // 64 N-groups per batch
  const int ng   = blockIdx.x & 63;
  const int mBase = wid << 4;
  const int nBase = ng << 6;
  const int m  = lane & 15;
  const int hi = lane >> 4;
  const int kb = hi << 1;

  v8f accF[4], accG[4];
#pragma unroll
  for (int nt = 0; nt < 4; ++nt) {
    accF[nt] = (v8f)0.0f;
    accG[nt] = (v8f)0.0f;
  }

  const float* ftB = Ft  + (size_t)b * CIN * HW + nBase;
  const float* feB = Fte + (size_t)b * CIN * HW + nBase;
  const float* wfRow = Wf + (mBase + m) * CIN;
  const float* wgRow = Wg + (mBase + m) * CIN;

  for (int kc0 = 0; kc0 < CIN; kc0 += KC) {
    __syncthreads();                               // protect previous reads
    // Stage B panels: [matrix][pixel][k], k-contiguous for b64 WMMA fetches.
    // Global side: 512 consecutive threads sweep 64-float rows (coalesced).
#pragma unroll
    for (int i = 0; i < (2 * KC * 64) / 512; ++i) {
      const int idx = threadIdx.x + i * 512;
      const int m2  = idx >> 11;                   // 0: Ft, 1: Fte
      const int k   = (idx >> 6) & (KC - 1);
      const int p   = idx & 63;
      const float* src = (m2 ? feB : ftB) + (size_t)(kc0 + k) * HW + p;
      lds_copy_b32(src, &bshare[(m2 * 64 + p) * KPAD + k]);
    }
    lds_copy_fence();
    __syncthreads();

#pragma unroll
    for (int k0 = 0; k0 < KC; k0 += 4) {
      const int kr = k0 + kb;                      // even -> 8B aligned
      const v2f aF = *reinterpret_cast<const v2f*>(wfRow + kc0 + kr);
      const v2f aG = *reinterpret_cast<const v2f*>(wgRow + kc0 + kr);
#pragma unroll
      for (int nt = 0; nt < 4; ++nt) {
        const int p = nt * 16 + m;
        const v2f bF = *reinterpret_cast<const v2f*>(&bshare[p * KPAD + kr]);
        const v2f bG = *reinterpret_cast<const v2f*>(&bshare[(64 + p) * KPAD + kr]);
        accF[nt] = __builtin_amdgcn_wmma_f32_16x16x4_f32(
            false, aF, false, bF, (short)0, accF[nt], false, false);
        accG[nt] = __builtin_amdgcn_wmma_f32_16x16x4_f32(
            false, aG, false, bG, (short)0, accG[nt], false, false);
      }
    }
  }

#pragma unroll
  for (int r = 0; r < 8; ++r) {
    const int oc  = mBase + r + 8 * hi;
    const float vbf = bf[oc];
    const float vbg = bg[oc];
    const size_t rowOff = ((size_t)b * CIN + oc) * HW;
#pragma unroll
    for (int nt = 0; nt < 4; ++nt) {
      const int n = nBase + nt * 16 + m;
      Ef[rowOff + n] = accF[nt][r] + vbf;
      Eg[rowOff + n] = accG[nt][r] + vbg;
    }
  }
}

// ---------------------------------------------------------------------------
// Kernel 2: local attention. 64 threads own an 8x8 pixel tile; halo 16x16.
// LDS layout: tile[pix * PSTR + c], PSTR=20 so float4 reads are 16B aligned.
// ---------------------------------------------------------------------------
#define TS   8
#define HALO 4
#define HT   16                   // TS + 2*HALO
#define CH   16                   // channel chunk
#define PSTR 20                   // padded per-pixel stride (floats)

__global__ __launch_bounds__(64) void psla_attn(
    const float* __restrict__ Ft, const float* __restrict__ Ef,
    const float* __restrict__ Eg, float* __restrict__ Out) {
  // offset table: (0,0) + rings s=1..4 (8 each). Order is irrelevant
  // (softmax + weighted sum are permutation invariant over k).
  const int OX[NOFF] = {0, -1,-1,-1, 0, 0, 1, 1, 1,
                           -2,-2,-2, 0, 0, 2, 2, 2,
                           -3,-3,-3, 0, 0, 3, 3, 3,
                           -4,-4,-4, 0, 0, 4, 4, 4};
  const int OY[NOFF] = {0, -1, 0, 1,-1, 1,-1, 0, 1,
                           -2, 0, 2,-2, 2,-2, 0, 2,
                           -3, 0, 3,-3, 3,-3, 0, 3,
                           -4, 0, 4,-4, 4,-4, 0, 4};

  __shared__ float tile[HT * HT * PSTR];   // 20480 B

  const int b  = blockIdx.z;
  const int tx = blockIdx.x * TS;          // w origin of tile
  const int ty = blockIdx.y * TS;          // h origin of tile
  const int lx = threadIdx.x & 7;
  const int ly = threadIdx.x >> 3;
  const int w  = tx + lx;
  const int h  = ty + ly;
  const size_t base = (size_t)b * CIN * HW;
  const int pixG = h * WW + w;

  float aff[NOFF];
#pragma unroll
  for (int k = 0; k < NOFF; ++k) aff[k] = 0.0f;

  // ---- pass 1: affinities ----
  for (int c0 = 0; c0 < CIN; c0 += CH) {
    __syncthreads();   // protect previous chunk's reads
    for (int idx = threadIdx.x; idx < HT * HT * CH; idx += 64) {
      const int c = idx >> 8;               // HT*HT == 256
      const int p = idx & 255;
      int hy = ty - HALO + (p >> 4);
      int hx = tx - HALO + (p & 15);
      hy = min(max(hy, 0), HH - 1);
      hx = min(max(hx, 0), WW - 1);
      lds_copy_b32(&Ef[base + (size_t)(c0 + c) * HW + hy * WW + hx],
                   &tile[p * PSTR + c]);
    }
    lds_copy_fence();
    __syncthreads();

    float eg[CH];
#pragma unroll
    for (int c = 0; c < CH; ++c)
      eg[c] = Eg[base + (size_t)(c0 + c) * HW + pixG];

#pragma unroll
    for (int k = 0; k < NOFF; ++k) {
      const int p = (ly + HALO + OX[k]) * HT + (lx + HALO + OY[k]);
      const float* t = &tile[p * PSTR];
      float s = 0.0f;
#pragma unroll
      for (int c = 0; c < CH; c += 4) {
        const float4 v = *reinterpret_cast<const float4*>(t + c);
        s = fmaf(eg[c + 0], v.x, s);
        s = fmaf(eg[c + 1], v.y, s);
        s = fmaf(eg[c + 2], v.z, s);
        s = fmaf(eg[c + 3], v.w, s);
      }
      aff[k] += s;
    }
  }

  // ---- masked softmax over k (k=0 is always valid) ----
  bool valid[NOFF];
  float mx = -INFINITY;
#pragma unroll
  for (int k = 0; k < NOFF; ++k) {
    const int hh = h + OX[k], ww2 = w + OY[k];
    valid[k] = (hh >= 0) & (hh < HH) & (ww2 >= 0) & (ww2 < WW);
    if (valid[k]) mx = fmaxf(mx, aff[k]);
  }
  float sum = 0.0f;
#pragma unroll
  for (int k = 0; k < NOFF; ++k) {
    const float e = valid[k] ? __expf(aff[k] - mx) : 0.0f;
    aff[k] = e;
    sum += e;
  }
  const float inv = 1.0f / sum;
#pragma unroll
  for (int k = 0; k < NOFF; ++k) aff[k] *= inv;

  // ---- pass 2: weighted gather of Ft neighbors ----
  for (int c0 = 0; c0 < CIN; c0 += CH) {
    __syncthreads();
    for (int idx = threadIdx.x; idx < HT * HT * CH; idx += 64) {
      const int c = idx >> 8;
      const int p = idx & 255;
      int hy = ty - HALO + (p >> 4);
      int hx = tx - HALO + (p & 15);
      hy = min(max(hy, 0), HH - 1);
      hx = min(max(hx, 0), WW - 1);
      lds_copy_b32(&Ft[base + (size_t)(c0 + c) * HW + hy * WW + hx],
                   &tile[p * PSTR + c]);
    }
    lds_copy_fence();
    __syncthreads();

#pragma unroll
    for (int c = 0; c < CH; c += 4) {
      float ax = 0.0f, ay = 0.0f, az = 0.0f, aw = 0.0f;
#pragma unroll
      for (int k = 0; k < NOFF; ++k) {
        const int p = (ly + HALO + OX[k]) * HT + (lx + HALO + OY[k]);
        const float4 v = *reinterpret_cast<const float4*>(&tile[p * PSTR + c]);
        ax = fmaf(aff[k], v.x, ax);
        ay = fmaf(aff[k], v.y, ay);
        az = fmaf(aff[k], v.z, az);
        aw = fmaf(aff[k], v.w, aw);
      }
      Out[base + (size_t)(c0 + c + 0) * HW + pixG] = ax;
      Out[base + (size_t)(c0 + c + 1) * HW + pixG] = ay;
      Out[base + (size_t)(c0 + c + 2) * HW + pixG] = az;
      Out[base + (size_t)(c0 + c + 3) * HW + pixG] = aw;
    }
  }
}

// ---------------------------------------------------------------------------
extern "C" void kernel_launch(void* const* d_in, const int* in_sizes, int n_in,
                              void* d_out, int out_size, void* d_ws, size_t ws_size,
                              hipStream_t stream) {
  const float* Ft  = (const float*)d_in[0];
  const float* Fte = (const float*)d_in[1];
  const float* Wf  = (const float*)d_in[2];
  const float* bf  = (const float*)d_in[3];
  const float* Wg  = (const float*)d_in[4];
  const float* bg  = (const float*)d_in[5];
  float* Out = (float*)d_out;

  // workspace: Ef then Eg, 8 MiB each (needs >= 16 MiB of d_ws)
  float* Ef = (float*)d_ws;
  float* Eg = Ef + (size_t)BATCH * CIN * HW;

  // 128 blocks x 16 waves = 2048 waves (16 M-tiles x 64 N-groups x 2 batches)
  psla_conv1x1_wmma<<<BATCH * 64, 512, 0, stream>>>(Ft, Fte, Wf, bf, Wg, bg, Ef, Eg);

  dim3 grid(WW / TS, HH / TS, BATCH);   // 8 x 8 x 2 = 128 blocks
  psla_attn<<<grid, 64, 0, stream>>>(Ft, Ef, Eg, Out);
}